// TRAdaptiveWaveletDecomposition_80539226735379
// MI455X (gfx1250) — compile-verified
//
#include <hip/hip_runtime.h>
#include <hip/hip_bf16.h>

// CDNA5 / gfx1250: fp32 CWT correlation recast as Toeplitz GEMM on
// V_WMMA_F32_16X16X4_F32 (wave32, D[16x16] = A[16x4] x B[4x16] + C).
// Each wave keeps two accumulator tiles sharing the B fragment.

typedef float v2f __attribute__((ext_vector_type(2)));
typedef float v8f __attribute__((ext_vector_type(8)));

#define NPSI_C   1024
#define LMAX_C   544
#define B_C      8
#define R_C      400
#define T_C      1200
#define S_C      5
#define BS_C     (B_C * S_C)   // 40
#define FP_ROWS  560           // Toeplitz rows: covers L+18 rounded to 4, L<=544
#define PADL_C   272           // left zero pad: min gi = start-544 >= -271
#define NGP_C    1765          // padded row stride (odd mod 64 -> no bank conflicts)
                               // max padded idx (tbase<=1200) = 1762 < 1765
#define FSZ_C    (FP_ROWS * 16)
#define COEF_ELEMS (8LL * 400 * 5 * 1200)

// ---------------------------------------------------------------------------
// Kernel 1: INT_PSI[i] = (cumsum of exp(-x^2/2)*cos(5x)) * STEP, double math
// (matches numpy float64 cumsum then float32 cast).
// ---------------------------------------------------------------------------
__global__ void psi_kernel(float* __restrict__ int_psi) {
  if (threadIdx.x == 0 && blockIdx.x == 0) {
    const double step = 16.0 / 1023.0;
    double acc = 0.0;
    for (int i = 0; i < NPSI_C; ++i) {
      double x = -8.0 + step * (double)i;
      acc += exp(-x * x * 0.5) * cos(5.0 * x);
      int_psi[i] = (float)(acc * step);
    }
  }
}

// ---------------------------------------------------------------------------
// Kernel 2: per-(b,s) wavelet kernel K[544], mask length L, slice start,
// K-loop trip P4, sqrt(scale); also the frequency_bands output tail.
// ---------------------------------------------------------------------------
__global__ void build_kernel(const float* __restrict__ int_psi,
                             const float* __restrict__ tr_values,
                             const float* __restrict__ lsp,
                             const float* __restrict__ bwp,
                             float* __restrict__ Kbuf,
                             float* __restrict__ meta,
                             float* __restrict__ fbands) {
  int id = threadIdx.x;
  if (id >= BS_C) return;
  int b = id / S_C, s = id % S_C;
  float trv   = tr_values[b];
  float scale = expf(lsp[s]) / (trv * 0.5f);          // exp(ls)/(tr/REF_TR)
  const float STEPF = (float)(16.0 / 1023.0);
  float denom = scale * STEPF;
  int   L0    = (int)floorf(scale * 16.0f) + 2;

  float* K = Kbuf + id * LMAX_C;
  int L = 0;
  for (int i = 0; i < LMAX_C; ++i) {
    int j = (int)floorf((float)i / denom);
    bool m = (i < L0) && (j < NPSI_C);
    int jc = j < 0 ? 0 : (j > NPSI_C - 1 ? NPSI_C - 1 : j);
    K[i] = m ? int_psi[jc] : 0.0f;
    L += m ? 1 : 0;
  }
  int start = LMAX_C + (L - 2) / 2 + 2 - L;   // L >= 2 always here
  int P4    = ((L + 18) >> 2) << 2;           // covers p in [0, L+15)

  float* mt = meta + id * 8;
  mt[0] = sqrtf(scale);
  mt[1] = __int_as_float(start);
  mt[2] = __int_as_float(P4);
  mt[3] = __int_as_float(L);

  float center = 1.0f / (scale * trv);
  float bw = bwp[s];
  fbands[id * 2 + 0] = fmaxf(0.008f, center * (1.0f - bw * 0.5f));
  fbands[id * 2 + 1] = fminf(0.12f,  center * (1.0f + bw * 0.5f));
}

// ---------------------------------------------------------------------------
// Kernel 3: main CWT. grid = (25 r-tiles, 40 (b,s)); block = 256 = 8 wave32.
// Single shared array forces the layout: F (Toeplitz filter) at offset 0 so
// its DS addresses fit immediate offsets, then zero-padded signal rows G.
// Branch-free inner loop; each wave runs two tiles (tt0, tt0+1) that share
// each B fragment: 3 LDS dwords per WMMA, two independent acc chains.
// ---------------------------------------------------------------------------
__global__ __launch_bounds__(256) void cwt_kernel(
    const float* __restrict__ ts,
    const float* __restrict__ Kbuf,
    const float* __restrict__ meta,
    float* __restrict__ out) {
  __shared__ float smem[FSZ_C + 16 * NGP_C];   // 35,840 + 112,960 = 148,800 B
  float* F = smem;
  float* G = smem + FSZ_C;

  const int bs    = blockIdx.y;
  const int b     = bs / S_C, s = bs % S_C;
  const int rbase = blockIdx.x * 16;
  const int tid   = threadIdx.x;

  const float* mt = meta + bs * 8;
  const float sqs   = mt[0];
  const int   start = __float_as_int(mt[1]);
  const int   P4    = __float_as_int(mt[2]);
  const int   L     = __float_as_int(mt[3]);

  // Banded Toeplitz filter matrix F[p][j] = K[p-j] (0 outside band)
  const float* K = Kbuf + bs * LMAX_C;
  for (int idx = tid; idx < FSZ_C; idx += 256) {
    int p = idx >> 4;
    int j = idx & 15;
    int d = p - j;
    F[idx] = (d >= 0 && d < L) ? K[d] : 0.0f;
  }
  // Zero-padded g rows: g[0]=x[0], g[i]=x[i]-x[i-1], g[T]=-x[T-1], 0 outside
  const float* tsb = ts + (size_t)(b * R_C + rbase) * T_C;
  for (int idx = tid; idx < 16 * NGP_C; idx += 256) {
    int rr = idx / NGP_C;
    int cc = idx - rr * NGP_C - PADL_C;   // g-index
    const float* row = tsb + (size_t)rr * T_C;
    float v = 0.0f;
    if (cc == 0)                    v = row[0];
    else if (cc == T_C)             v = -row[T_C - 1];
    else if (cc > 0 && cc < T_C)    v = row[cc] - row[cc - 1];
    G[idx] = v;
  }
  __syncthreads();

  const int wave = tid >> 5;
  const int lane = tid & 31;
  const int half = lane >> 4;   // K-pair select (ISA A/B fragment layout)
  const int lid  = lane & 15;   // M (rows) for A, N (cols) for B/D
  const float* grow = G + lid * NGP_C;
  const float* fcol = F + half * 32 + lid;   // + p0*16 per step

  // Pairs (tt0, tt0+1): tt0 = wave*2 + 16k covers all even tiles 0..74.
  for (int tt0 = wave * 2; tt0 < 75; tt0 += 16) {
    const int tb0 = tt0 * 16;
    v8f acc0 = {0.f, 0.f, 0.f, 0.f, 0.f, 0.f, 0.f, 0.f};
    v8f acc1 = {0.f, 0.f, 0.f, 0.f, 0.f, 0.f, 0.f, 0.f};
    // padded g index of A[r, p]: start + tbase + p - 544 + PADL
    const int cb0 = start + tb0 - LMAX_C + PADL_C + half * 2;
    #pragma unroll 4
    for (int p0 = 0; p0 < P4; p0 += 4) {
      int gi = cb0 + p0;
      v2f a0 = {grow[gi],      grow[gi + 1]};
      v2f a1 = {grow[gi + 16], grow[gi + 17]};   // tile tt0+1: tbase += 16
      const float* fp = fcol + p0 * 16;
      v2f bf = {fp[0], fp[16]};                  // shared B fragment
      // (neg_a, A, neg_b, B, c_mod, C, reuse_a, reuse_b)
      acc0 = __builtin_amdgcn_wmma_f32_16x16x4_f32(
          false, a0, false, bf, (short)0, acc0, false, false);
      acc1 = __builtin_amdgcn_wmma_f32_16x16x4_f32(
          false, a1, false, bf, (short)0, acc1, false, false);
    }
    // D layout: lane N=lid, VGPR v -> M = v + 8*half
    const int t0 = tb0 + lid;
    #pragma unroll
    for (int v = 0; v < 8; ++v) {
      int r = rbase + v + half * 8;
      out[((size_t)(b * R_C + r) * S_C + s) * T_C + t0] = sqs * fabsf(acc0[v]);
    }
    if (tt0 + 1 < 75) {               // wave-uniform guard, outside hot loop
      const int t1 = tb0 + 16 + lid;
      #pragma unroll
      for (int v = 0; v < 8; ++v) {
        int r = rbase + v + half * 8;
        out[((size_t)(b * R_C + r) * S_C + s) * T_C + t1] = sqs * fabsf(acc1[v]);
      }
    }
  }
}

// ---------------------------------------------------------------------------
extern "C" void kernel_launch(void* const* d_in, const int* in_sizes, int n_in,
                              void* d_out, int out_size, void* d_ws, size_t ws_size,
                              hipStream_t stream) {
  const float* ts  = (const float*)d_in[0];  // [8,400,1200] f32
  const float* tr  = (const float*)d_in[1];  // [8] f32
  const float* lsp = (const float*)d_in[2];  // [5] f32
  const float* bwp = (const float*)d_in[3];  // [5] f32
  float* out = (float*)d_out;                // 19,200,000 coeffs + 80 bands
  float* ws  = (float*)d_ws;

  float* int_psi = ws;                       // 1024 f32
  float* Kbuf    = ws + 1024;                // 40 * 544 f32
  float* meta    = ws + 1024 + BS_C * LMAX_C;// 40 * 8 f32

  psi_kernel<<<1, 1, 0, stream>>>(int_psi);
  build_kernel<<<1, 64, 0, stream>>>(int_psi, tr, lsp, bwp, Kbuf, meta,
                                     out + COEF_ELEMS);
  dim3 grid(R_C / 16, BS_C);                 // (25, 40)
  cwt_kernel<<<grid, 256, 0, stream>>>(ts, Kbuf, meta, out);
}